// Generator_Decoder_34832184770755
// MI455X (gfx1250) — compile-verified
//
#include <hip/hip_runtime.h>
#include <hip/hip_bf16.h>
#include <math.h>

typedef float v2f __attribute__((ext_vector_type(2)));
typedef float v8f __attribute__((ext_vector_type(8)));
typedef unsigned int u32x4 __attribute__((ext_vector_type(4)));
typedef int i32x8 __attribute__((ext_vector_type(8)));
typedef int i32x4 __attribute__((ext_vector_type(4)));

// ---------------- problem dimensions ----------------
#define BG    4096                 // batch of graphs
#define VV    9                    // nodes per graph
#define NNODE (BG*VV)              // 36864
#define NEDGE (BG*VV*VV)           // 331776 (with self loops)
#define ENS   (BG*VV*(VV-1))       // 294912 (self loops removed)

// output layout (elements): x | eattr | edge_index | batch
#define OFF_E   (NNODE*10)              // 368640
#define OFF_IDX (OFF_E + ENS*5)         // 1843200
#define OFF_B   (OFF_IDX + 2*ENS)       // 2433024

// =====================================================================
// 1-D Tensor Data Mover copy: global -> LDS, nelem f32 (contiguous).
// D# per CDNA5 ISA §8.3/8.4: group0 = {count=1, lds_addr, global_addr,
// type=2}; group1 = {data_size=4B, tensor_dim0 = tile_dim0 = nelem, 1-D}.
// Tracked on TENSORcnt; runs concurrently with subsequent ALU/WMMA work.
// =====================================================================
__device__ __forceinline__ void tdm_load_1d(const float* gsrc, float* lds_dst, int nelem)
{
  const unsigned lds_addr = (unsigned)(size_t)(void*)lds_dst;  // LDS generic addr: low 32 bits = LDS offset
  const unsigned long long ga = (unsigned long long)(size_t)gsrc;
  u32x4 g0;
  g0.x = 1u;                                                   // count=1, user descriptor
  g0.y = lds_addr;                                             // bits 63:32 lds_addr
  g0.z = (unsigned)ga;                                         // global_addr[31:0]
  g0.w = (unsigned)((ga >> 32) & 0x01FFFFFFu) | (2u << 30);    // addr[56:32] | type=2
  i32x8 g1;
  g1[0] = 0x00020000;                                          // data_size=2 (4 bytes)
  g1[1] = (int)((nelem & 0xFFFF) << 16);                       // tensor_dim0[15:0]
  g1[2] = (int)(((unsigned)nelem >> 16) | (1u << 16));         // tensor_dim0[31:16] | tensor_dim1=1
  g1[3] = (int)((nelem & 0xFFFF) << 16);                       // tile_dim0 (<=65535)
  g1[4] = 0;                                                   // tile_dim1=0 (1-D), tile_dim2=0
  g1[5] = nelem;                                               // tensor_dim0_stride[31:0]
  g1[6] = 0;
  g1[7] = 0;
  const i32x4 z4 = {0, 0, 0, 0};
  const i32x8 z8 = {0, 0, 0, 0, 0, 0, 0, 0};
  __builtin_amdgcn_tensor_load_to_lds(g0, g1, z4, z4, z8, 0);
}

// =====================================================================
// Generic f32 WMMA GEMM with TRANSPOSED B:
//   C[M,N] = act( A[M,K] @ B[K,N] + bias [+ C if accflag] ),  Bt stored [N][K]
// Requirements: M % 32 == 0, K % 4 == 0, lda even, K even.
// amode: 0 = direct rows, 1 = row -> SRC(edge), 2 = row -> DST(edge)
// act:   0 = none, 1 = tanh, 2 = leaky_relu(0.01)
// 128 threads = 4 waves; each wave computes a 32x32 block (2x2 WMMA tiles):
// inner loop is 4 contiguous b64 loads + 4 WMMAs, fully branch-free.
// =====================================================================
__global__ __launch_bounds__(128) void gemm_f32_wmma(
    const float* __restrict__ A, const float* __restrict__ Bt,
    const float* __restrict__ bias, float* __restrict__ C,
    int M, int N, int K, int lda, int ldc,
    int amode, int act, int accflag)
{
  const int lane = threadIdx.x & 31;
  const int wave = threadIdx.x >> 5;
  const int m0 = blockIdx.y << 5;                // 32 rows per block
  const int n0 = blockIdx.x * 128 + wave * 32;   // wave strip [n0, n0+32)
  if (n0 >= N) return;                           // wave-uniform exit

  const int l15  = lane & 15;
  const int kofs = (lane >> 4) << 1;             // lanes 0-15: K+0/+1, 16-31: K+2/+3
  int ar0 = m0 + l15;
  int ar1 = m0 + 16 + l15;
  if (amode) {
    int b = ar0 / 81, r = ar0 - b * 81, i = r / 9, j = r - i * 9;
    ar0 = (amode == 1) ? (b * 9 + i) : (b * 9 + j);
    b = ar1 / 81; r = ar1 - b * 81; i = r / 9; j = r - i * 9;
    ar1 = (amode == 1) ? (b * 9 + i) : (b * 9 + j);
  }
  const float* A0 = A + (size_t)ar0 * lda + kofs;
  const float* A1 = A + (size_t)ar1 * lda + kofs;
  // clamp OOB columns for addressing only; stores are guarded
  const int c0 = min(n0 + l15, N - 1);
  const int c1 = min(n0 + 16 + l15, N - 1);
  const float* B0 = Bt + (size_t)c0 * K + kofs;
  const float* B1 = Bt + (size_t)c1 * K + kofs;

  v8f acc00 = {0.f,0.f,0.f,0.f,0.f,0.f,0.f,0.f};
  v8f acc01 = acc00, acc10 = acc00, acc11 = acc00;
#pragma unroll 4
  for (int kk = 0; kk < K; kk += 4) {
    const v2f a0 = *(const v2f*)(A0 + kk);
    const v2f a1 = *(const v2f*)(A1 + kk);
    const v2f b0 = *(const v2f*)(B0 + kk);
    const v2f b1 = *(const v2f*)(B1 + kk);
    acc00 = __builtin_amdgcn_wmma_f32_16x16x4_f32(false, a0, false, b0, (short)0, acc00, false, false);
    acc01 = __builtin_amdgcn_wmma_f32_16x16x4_f32(false, a0, false, b1, (short)0, acc01, false, false);
    acc10 = __builtin_amdgcn_wmma_f32_16x16x4_f32(false, a1, false, b0, (short)0, acc10, false, false);
    acc11 = __builtin_amdgcn_wmma_f32_16x16x4_f32(false, a1, false, b1, (short)0, acc11, false, false);
  }

  const int rofs = (lane >> 4) << 3;
#pragma unroll
  for (int mt = 0; mt < 2; ++mt) {
#pragma unroll
    for (int nt = 0; nt < 2; ++nt) {
      const int col = n0 + nt * 16 + l15;
      if (col >= N) continue;
      const v8f acc = mt ? (nt ? acc11 : acc10) : (nt ? acc01 : acc00);
      const int rbase = m0 + mt * 16 + rofs;
      const float bv = bias ? bias[col] : 0.f;
#pragma unroll
      for (int r = 0; r < 8; ++r) {
        const size_t idx = (size_t)(rbase + r) * ldc + col;
        float v = acc[r] + bv;
        if (accflag) v += C[idx];
        if (act == 1)      v = tanhf(v);
        else if (act == 2) v = (v > 0.f) ? v : 0.01f * v;
        C[idx] = v;
      }
    }
  }
}

// =====================================================================
// Per-graph edge-conditioned multi-head attention.
// One block = one graph (9 nodes, 81 edges); 128 threads = 4 waves = 4 heads.
// k/v slices (contiguous 4.5KB each) staged into LDS by the Tensor Data
// Mover, overlapped with the e = eattr @ we projection (WMMA into LDS).
// KE = padded edge-feature count (8 layer 0, 32 after); wet stored [128][KE].
// =====================================================================
template <int KE>
__global__ __launch_bounds__(128) void attn_kernel(
    const float* __restrict__ q, const float* __restrict__ k,
    const float* __restrict__ v, const float* __restrict__ eattr,
    const float* __restrict__ wet, float* __restrict__ xout)
{
  __shared__ float sk[9][128];
  __shared__ float sv[9][128];
  __shared__ float se[96][128];
  __shared__ float slog[4][81];
  __shared__ float sout[4][9][32];

  const int g = blockIdx.x;
  const int tid = threadIdx.x;
  const int lane = tid & 31;
  const int wave = tid >> 5;

  // ---- async k/v staging via TDM (overlaps with WMMA below) ----
  const int wave_u = __builtin_amdgcn_readfirstlane(wave);
  if (wave_u == 0) {
    tdm_load_1d(k + (size_t)g * 9 * 128, &sk[0][0], 9 * 128);
    tdm_load_1d(v + (size_t)g * 9 * 128, &sv[0][0], 9 * 128);
  }

  // ---- e projection via WMMA: 6 Mtiles x 8 Ntiles, 4 waves round-robin ----
  {
    const int l15  = lane & 15;
    const int kofs = (lane >> 4) << 1;
    for (int t = wave; t < 48; t += 4) {
      const int mt = t >> 3, nt = t & 7;
      const int ml = min(mt * 16 + l15, 80);     // clamp: rows 81..95 unused
      const float* Ar = eattr + (size_t)(g * 81 + ml) * KE + kofs;
      const int bc = nt * 16 + l15;
      const float* Bp = wet + (size_t)bc * KE + kofs;
      v8f acc = {0.f,0.f,0.f,0.f,0.f,0.f,0.f,0.f};
#pragma unroll
      for (int kk = 0; kk < KE; kk += 4) {
        const v2f a  = *(const v2f*)(Ar + kk);
        const v2f bb = *(const v2f*)(Bp + kk);
        acc = __builtin_amdgcn_wmma_f32_16x16x4_f32(false, a, false, bb,
                                                    (short)0, acc, false, false);
      }
      const int rb = mt * 16 + ((lane >> 4) << 3);
#pragma unroll
      for (int r = 0; r < 8; ++r) se[rb + r][bc] = acc[r];
    }
  }
  __builtin_amdgcn_s_wait_tensorcnt(0);   // k/v TDM complete (no-op for waves 1-3)
  __syncthreads();

  const int h = wave;                       // head
  const float scale = 0.1767766952966369f;  // 1/sqrt(32)

  // logits[e] = (q[dst] . (k[src] + e)) / sqrt(D)
  for (int e = lane; e < 81; e += 32) {
    const int i = e / 9, j = e - i * 9;     // src=i, dst=j
    const float* qrow = q + (size_t)(g * 9 + j) * 128 + h * 32;
    float acc = 0.f;
#pragma unroll
    for (int d = 0; d < 32; ++d)
      acc += qrow[d] * (sk[i][h * 32 + d] + se[e][h * 32 + d]);
    slog[h][e] = acc * scale;
  }
  __syncthreads();

  // softmax over the 9 sources of each destination (lanes 0..8 = dst)
  if (lane < 9) {
    float mx = -3.402823e38f;
#pragma unroll
    for (int i = 0; i < 9; ++i) mx = fmaxf(mx, slog[h][i * 9 + lane]);
    float ex[9]; float s = 0.f;
#pragma unroll
    for (int i = 0; i < 9; ++i) { ex[i] = __expf(slog[h][i * 9 + lane] - mx); s += ex[i]; }
    const float r = 1.f / (s + 1e-16f);
#pragma unroll
    for (int i = 0; i < 9; ++i) slog[h][i * 9 + lane] = ex[i] * r;  // reuse as alpha
  }
  __syncthreads();

  // out[dst] = sum_src alpha * (v[src] + e)   (lane = dim)
  for (int j = 0; j < 9; ++j) {
    float acc = 0.f;
#pragma unroll
    for (int i = 0; i < 9; ++i) {
      const int e = i * 9 + j;
      acc += slog[h][e] * (sv[i][h * 32 + lane] + se[e][h * 32 + lane]);
    }
    sout[h][j][lane] = acc;
  }
  __syncthreads();

  for (int t = tid; t < 9 * 32; t += 128) {
    const int j = t >> 5, d = t & 31;
    xout[(size_t)(g * 9 + j) * 32 + d] =
        0.25f * (sout[0][j][d] + sout[1][j][d] + sout[2][j][d] + sout[3][j][d]);
  }
}

// =====================================================================
// LayerNorm(32) + leaky_relu ; one wave per row (wave32 shuffle reduce)
// =====================================================================
__global__ __launch_bounds__(256) void ln_leaky_kernel(
    const float* __restrict__ x, const float* __restrict__ gamma,
    const float* __restrict__ beta, float* __restrict__ y, int nrows)
{
  const int lane = threadIdx.x & 31;
  const int row = blockIdx.x * (blockDim.x >> 5) + (threadIdx.x >> 5);
  if (row >= nrows) return;
  const float v = x[(size_t)row * 32 + lane];
  float s = v;
  for (int o = 16; o; o >>= 1) s += __shfl_xor(s, o, 32);
  const float mu = s * (1.f / 32.f);
  const float d = v - mu;
  float qq = d * d;
  for (int o = 16; o; o >>= 1) qq += __shfl_xor(qq, o, 32);
  float out = d * rsqrtf(qq * (1.f / 32.f) + 1e-5f) * gamma[lane] + beta[lane];
  y[(size_t)row * 32 + lane] = out > 0.f ? out : 0.01f * out;
}

// =====================================================================
// transpose + zero-pad rows: out[Cols][Rpad] <- in[Rin, Cols]
// =====================================================================
__global__ void transpose_pad_kernel(const float* __restrict__ in, float* __restrict__ out,
                                     int Rin, int Rpad, int Cols)
{
  const int t = blockIdx.x * blockDim.x + threadIdx.x;
  if (t >= Rpad * Cols) return;
  const int r = t / Cols, c = t - r * Cols;
  out[(size_t)c * Rpad + r] = (r < Rin) ? in[(size_t)r * Cols + c] : 0.f;
}

// scatter node logits -> x0 [N,16] (pad 13->16)
__global__ void build_x0_kernel(const float* __restrict__ nl, float* __restrict__ x0)
{
  const int t = blockIdx.x * blockDim.x + threadIdx.x;
  if (t >= NNODE * 16) return;
  const int n = t >> 4, c = t & 15;
  const int b = n / 9, vv = n - b * 9;
  x0[t] = (c < 13) ? nl[(size_t)b * 117 + vv * 13 + c] : 0.f;
}

// symmetrize edge logits + transpose -> eattr0 [E,8] (pad 5->8)
__global__ void build_e0_kernel(const float* __restrict__ el, float* __restrict__ e0)
{
  const int t = blockIdx.x * blockDim.x + threadIdx.x;
  if (t >= NEDGE * 8) return;
  const int e = t >> 3, c = t & 7;
  const int b = e / 81, r = e - b * 81;
  const int i = r / 9, j = r - i * 9;
  float val = 0.f;
  if (c < 5) {
    const float* base = el + (size_t)b * 405 + c * 81;
    val = 0.5f * (base[i * 9 + j] + base[j * 9 + i]);
  }
  e0[t] = val;
}

// =====================================================================
// remove_self_loops compaction of eattr, edge_index, batch
// =====================================================================
__global__ void finalize_eattr_kernel(const float* __restrict__ efin,
                                      float* __restrict__ out)
{
  const int t = blockIdx.x * blockDim.x + threadIdx.x;
  if (t >= ENS * 5) return;
  const int ec = t / 5, c = t - ec * 5;
  const int b = ec / 72, r = ec - b * 72;
  const int i = r >> 3, jj = r & 7;
  const int j = jj + (jj >= i ? 1 : 0);
  out[OFF_E + t] = efin[(size_t)(b * 81 + i * 9 + j) * 5 + c];
}

__global__ void finalize_idx_kernel(float* __restrict__ outf)
{
  int* out = (int*)outf;
  const int t = blockIdx.x * blockDim.x + threadIdx.x;
  if (t < ENS) {
    const int b = t / 72, r = t - b * 72;
    const int i = r >> 3, jj = r & 7;
    const int j = jj + (jj >= i ? 1 : 0);
    out[OFF_IDX + t]       = b * 9 + i;   // SRC
    out[OFF_IDX + ENS + t] = b * 9 + j;   // DST
  }
  if (t < NNODE) out[OFF_B + t] = t / 9;  // batch
}

// =====================================================================
// Host side
// =====================================================================
static inline void launch_gemm(hipStream_t s, const float* A, const float* Bt,
                               const float* bias, float* C,
                               int M, int N, int K, int lda, int ldc,
                               int amode, int act, int accflag)
{
  dim3 grid((unsigned)((N + 127) / 128), (unsigned)(M / 32), 1);
  gemm_f32_wmma<<<grid, dim3(128, 1, 1), 0, s>>>(A, Bt, bias, C, M, N, K,
                                                 lda, ldc, amode, act, accflag);
}

static void conv_layer(hipStream_t s,
                       const float* xin, int Kx,           // lda == Kx (padded)
                       const float* ein, int Ke,           // lde == Ke (padded)
                       const float* t_wq, const float* t_wk, const float* t_wv,
                       const float* t_we,
                       const float* t_ws, const float* t_wd, const float* t_wa,
                       float* q, float* k, float* v,
                       float* xattn, float* eout, int eact)
{
  launch_gemm(s, xin, t_wq, nullptr, q, NNODE, 128, Kx, Kx, 128, 0, 0, 0);
  launch_gemm(s, xin, t_wk, nullptr, k, NNODE, 128, Kx, Kx, 128, 0, 0, 0);
  launch_gemm(s, xin, t_wv, nullptr, v, NNODE, 128, Kx, Kx, 128, 0, 0, 0);
  if (Ke == 8) attn_kernel<8> <<<BG, 128, 0, s>>>(q, k, v, ein, t_we, xattn);
  else         attn_kernel<32><<<BG, 128, 0, s>>>(q, k, v, ein, t_we, xattn);
  // eout = act( x[SRC]@weo_s + x[DST]@weo_d + eattr@weo_e )
  launch_gemm(s, xin, t_ws, nullptr, eout, NEDGE, 32, Kx, Kx, 32, 1, 0,    0);
  launch_gemm(s, xin, t_wd, nullptr, eout, NEDGE, 32, Kx, Kx, 32, 2, 0,    1);
  launch_gemm(s, ein, t_wa, nullptr, eout, NEDGE, 32, Ke, Ke, 32, 0, eact, 1);
}

extern "C" void kernel_launch(void* const* d_in, const int* in_sizes, int n_in,
                              void* d_out, int out_size, void* d_ws, size_t ws_size,
                              hipStream_t stream)
{
  (void)in_sizes; (void)n_in; (void)out_size; (void)ws_size;
  const float* latent  = (const float*)d_in[0];
  const float* w0 = (const float*)d_in[1];  const float* b0 = (const float*)d_in[2];
  const float* w1 = (const float*)d_in[3];  const float* b1 = (const float*)d_in[4];
  const float* w2 = (const float*)d_in[5];  const float* b2 = (const float*)d_in[6];
  const float* w_edges = (const float*)d_in[7];  const float* b_edges = (const float*)d_in[8];
  const float* w_nodes = (const float*)d_in[9];  const float* b_nodes = (const float*)d_in[10];
  const float* c0_wq = (const float*)d_in[11]; const float* c0_wk = (const float*)d_in[12];
  const float* c0_wv = (const float*)d_in[13]; const float* c0_we = (const float*)d_in[14];
  const float* c0_weo = (const float*)d_in[15];
  const float* c1_wq = (const float*)d_in[16]; const float* c1_wk = (const float*)d_in[17];
  const float* c1_wv = (const float*)d_in[18]; const float* c1_we = (const float*)d_in[19];
  const float* c1_weo = (const float*)d_in[20];
  const float* ln0_g = (const float*)d_in[21]; const float* ln0_b = (const float*)d_in[22];
  const float* ln1_g = (const float*)d_in[23]; const float* ln1_b = (const float*)d_in[24];
  const float* w_feat = (const float*)d_in[25]; const float* w_eout = (const float*)d_in[26];

  float* ws = (float*)d_ws;
  size_t off = 0;
  auto take = [&](size_t n) { float* p = ws + off; off += n; return p; };
  float* h0    = take((size_t)BG * 128);
  float* h1    = take((size_t)BG * 256);
  float* h2    = take((size_t)BG * 512);
  float* nlraw = take((size_t)BG * 117 + 3);
  float* elraw = take((size_t)BG * 405 + 3);
  float* x0    = take((size_t)NNODE * 16);
  float* e0    = take((size_t)NEDGE * 8);
  float* q     = take((size_t)NNODE * 128);
  float* k     = take((size_t)NNODE * 128);
  float* v     = take((size_t)NNODE * 128);
  float* xattn = take((size_t)NNODE * 32);
  float* xA    = take((size_t)NNODE * 32);
  float* xB    = take((size_t)NNODE * 32);
  float* eA    = take((size_t)NEDGE * 32);
  float* eB    = take((size_t)NEDGE * 32);
  float* efin  = take((size_t)NEDGE * 5 + 3);
  // transposed (and zero-padded where needed) weights, stored [N][K]
  float* t_w0   = take(128 * 128);
  float* t_w1   = take(256 * 128);
  float* t_w2   = take(512 * 256);
  float* t_wn   = take(117 * 512 + 2);
  float* t_wed  = take(405 * 512 + 2);
  float* t0_wq  = take(128 * 16);
  float* t0_wk  = take(128 * 16);
  float* t0_wv  = take(128 * 16);
  float* t0_we  = take(128 * 8);
  float* t0_ws  = take(32 * 16);
  float* t0_wd  = take(32 * 16);
  float* t0_wa  = take(32 * 8);
  float* t1_wq  = take(128 * 32);
  float* t1_wk  = take(128 * 32);
  float* t1_wv  = take(128 * 32);
  float* t1_we  = take(128 * 32);
  float* t1_ws  = take(32 * 32);
  float* t1_wd  = take(32 * 32);
  float* t1_wa  = take(32 * 32);
  float* t_wft  = take(10 * 32);
  float* t_weo  = take(5 * 32 + 2);

  hipStream_t s = stream;
  auto tp = [&](const float* in, float* out, int Rin, int Rpad, int Cols) {
    const int n = Rpad * Cols;
    transpose_pad_kernel<<<(n + 255) / 256, 256, 0, s>>>(in, out, Rin, Rpad, Cols);
  };
  tp(w0, t_w0, 128, 128, 128);
  tp(w1, t_w1, 128, 128, 256);
  tp(w2, t_w2, 256, 256, 512);
  tp(w_nodes, t_wn, 512, 512, 117);
  tp(w_edges, t_wed, 512, 512, 405);
  tp(c0_wq, t0_wq, 13, 16, 128);
  tp(c0_wk, t0_wk, 13, 16, 128);
  tp(c0_wv, t0_wv, 13, 16, 128);
  tp(c0_we, t0_we, 5, 8, 128);
  tp(c0_weo,           t0_ws, 13, 16, 32);
  tp(c0_weo + 13 * 32, t0_wd, 13, 16, 32);
  tp(c0_weo + 26 * 32, t0_wa,  5,  8, 32);
  tp(c1_wq, t1_wq, 32, 32, 128);
  tp(c1_wk, t1_wk, 32, 32, 128);
  tp(c1_wv, t1_wv, 32, 32, 128);
  tp(c1_we, t1_we, 32, 32, 128);
  tp(c1_weo,           t1_ws, 32, 32, 32);
  tp(c1_weo + 32 * 32, t1_wd, 32, 32, 32);
  tp(c1_weo + 64 * 32, t1_wa, 32, 32, 32);
  tp(w_feat, t_wft, 32, 32, 10);
  tp(w_eout, t_weo, 32, 32, 5);

  // ---- Generator2 MLP (tanh) ----
  launch_gemm(s, latent, t_w0, b0, h0, BG, 128, 128, 128, 128, 0, 1, 0);
  launch_gemm(s, h0,     t_w1, b1, h1, BG, 256, 128, 128, 256, 0, 1, 0);
  launch_gemm(s, h1,     t_w2, b2, h2, BG, 512, 256, 256, 512, 0, 1, 0);
  launch_gemm(s, h2, t_wn, b_nodes, nlraw, BG, 117, 512, 512, 117, 0, 0, 0);
  launch_gemm(s, h2, t_wed, b_edges, elraw, BG, 405, 512, 512, 405, 0, 0, 0);
  build_x0_kernel<<<(NNODE * 16 + 255) / 256, 256, 0, s>>>(nlraw, x0);
  build_e0_kernel<<<(NEDGE * 8 + 255) / 256, 256, 0, s>>>(elraw, e0);

  // ---- GNN layer 0 ----
  conv_layer(s, x0, 16, e0, 8, t0_wq, t0_wk, t0_wv, t0_we, t0_ws, t0_wd, t0_wa,
             q, k, v, xattn, eA, /*eact=*/2);
  ln_leaky_kernel<<<NNODE / 8, 256, 0, s>>>(xattn, ln0_g, ln0_b, xA, NNODE);

  // ---- GNN layer 1 ----
  conv_layer(s, xA, 32, eA, 32, t1_wq, t1_wk, t1_wv, t1_we, t1_ws, t1_wd, t1_wa,
             q, k, v, xattn, eB, /*eact=*/2);
  ln_leaky_kernel<<<NNODE / 8, 256, 0, s>>>(xattn, ln1_g, ln1_b, xB, NNODE);

  // ---- final conv (reuses layer-1 weights, reference bug reproduced) ----
  conv_layer(s, xB, 32, eB, 32, t1_wq, t1_wk, t1_wv, t1_we, t1_ws, t1_wd, t1_wa,
             q, k, v, xattn, eA, /*eact=*/0);

  // ---- outputs ----
  launch_gemm(s, xattn, t_wft, nullptr, (float*)d_out, NNODE, 10, 32, 32, 10, 0, 0, 0);
  launch_gemm(s, eA, t_weo, nullptr, efin, NEDGE, 5, 32, 32, 5, 0, 0, 0);
  finalize_eattr_kernel<<<(ENS * 5 + 255) / 256, 256, 0, s>>>(efin, (float*)d_out);
  finalize_idx_kernel<<<(ENS + 255) / 256, 256, 0, s>>>((float*)d_out);
}